// GCNLayer_67808943669323
// MI455X (gfx1250) — compile-verified
//
#include <hip/hip_runtime.h>

#define IN_DIM 64
#define OUT_DIM 64

typedef float v2f __attribute__((ext_vector_type(2)));
typedef float v8f __attribute__((ext_vector_type(8)));

// ---------------------------------------------------------------------------
// Phase 0: zero workspace (deg_out[N], deg_in[N], agg[N*64]) each call so the
// kernel is deterministic across graph replays.
// ---------------------------------------------------------------------------
__global__ void zero_ws_kernel(float* __restrict__ p, long n) {
    long i = (long)blockIdx.x * blockDim.x + threadIdx.x;
    if (i < n) p[i] = 0.0f;
}

// ---------------------------------------------------------------------------
// Phase 1: degree counts via float atomics (exact for counts < 2^24).
// edge_index is [2, E] row-major: src = row 0, dst = row 1.
// ---------------------------------------------------------------------------
__global__ void degree_kernel(const int* __restrict__ src, const int* __restrict__ dst,
                              float* __restrict__ deg_out, float* __restrict__ deg_in,
                              int E) {
    int e = blockIdx.x * blockDim.x + threadIdx.x;
    if (e < E) {
        atomicAdd(deg_out + src[e], 1.0f);
        atomicAdd(deg_in  + dst[e], 1.0f);
    }
}

// ---------------------------------------------------------------------------
// Phase 2: edge gather + normalized scatter-add.
// 16 lanes cooperate per edge; each lane moves one float4 (16*4 = 64 floats).
// norm = rsqrt(max(deg_out[src],1) * max(deg_in[dst],1)); the redundant scalar
// degree loads hit L0/L2 (x, agg, deg all L2-resident at 192 MB).
// ---------------------------------------------------------------------------
__global__ void scatter_kernel(const float* __restrict__ x,
                               const int* __restrict__ src, const int* __restrict__ dst,
                               const float* __restrict__ deg_out,
                               const float* __restrict__ deg_in,
                               float* __restrict__ agg, int E) {
    int t = blockIdx.x * blockDim.x + threadIdx.x;
    int e = t >> 4;
    if (e >= E) return;
    int lane = t & 15;
    int s = src[e];
    int d = dst[e];
    float no = fmaxf(deg_out[s], 1.0f);
    float ni = fmaxf(deg_in[d], 1.0f);
    float norm = __frsqrt_rn(no * ni);
    const float4 xv = *reinterpret_cast<const float4*>(x + (size_t)s * IN_DIM + lane * 4);
    float* o = agg + (size_t)d * IN_DIM + lane * 4;
    atomicAdd(o + 0, xv.x * norm);
    atomicAdd(o + 1, xv.y * norm);
    atomicAdd(o + 2, xv.z * norm);
    atomicAdd(o + 3, xv.w * norm);
}

// ---------------------------------------------------------------------------
// Phase 3: out = agg @ W + b via V_WMMA_F32_16X16X4_F32 (fp32 end-to-end).
// One wave32 computes a 16x64 strip: 4 accumulators (one per 16-col tile),
// so each A fragment is loaded once and reused by 4 WMMAs (4x less A traffic
// than one-tile-per-wave). W (16 KB) is staged in LDS once per block,
// pre-interleaved as WL[k/2][n][k&1] so a B fragment (K=kb,kb+1 at column n)
// is one aligned ds_load_b64.
// Fragment layouts per CDNA5 ISA:
//   A (16x4 f32, 2 VGPR): lanes 0-15 -> M=lane, K={0,1}; lanes 16-31 -> K={2,3}
//   B (4x16 f32, 2 VGPR): lanes 0-15 -> N=lane, K={0,1}; lanes 16-31 -> K={2,3}
//   C/D (16x16 f32, 8 VGPR): VGPR j -> M=j (lanes 0-15) / M=j+8 (lanes 16-31),
//                            N = lane & 15.
// ---------------------------------------------------------------------------
__global__ void gemm_wmma_kernel(const float* __restrict__ agg,
                                 const float* __restrict__ W,
                                 const float* __restrict__ bias,
                                 float* __restrict__ out,
                                 int N, int rowTiles) {
    // LDS copy of W, interleaved: WL[(k>>1)*128 + n*2 + (k&1)], 4096 floats.
    __shared__ float WL[IN_DIM * OUT_DIM];

    // Cooperative interleave-load: 256 threads x 16 floats. Thread handles
    // 8 (k-pair, n) slots: reads W[k][n], W[k+1][n] and stores them adjacent.
    {
        // 2048 k-pair*n slots; each thread fills 8 slots.
        for (int i = threadIdx.x; i < (IN_DIM / 2) * OUT_DIM; i += blockDim.x) {
            const int kp = i >> 6;          // k pair index: k = 2*kp
            const int n  = i & 63;
            const float w0 = W[(size_t)(2 * kp) * OUT_DIM + n];
            const float w1 = W[(size_t)(2 * kp + 1) * OUT_DIM + n];
            WL[kp * 128 + n * 2 + 0] = w0;
            WL[kp * 128 + n * 2 + 1] = w1;
        }
    }
    __syncthreads();

    const int wave = (int)((blockIdx.x * blockDim.x + threadIdx.x) >> 5);
    const int lane = threadIdx.x & 31;
    if (wave >= rowTiles) return;           // uniform per wave: EXEC stays all-1s

    const int rowBase = wave * 16;
    const int l15 = lane & 15;
    const int hi  = lane >> 4;              // 0: K pair {0,1}, 1: K pair {2,3}

    v8f c[4];
#pragma unroll
    for (int tc = 0; tc < 4; ++tc) {
        const float bn = bias[tc * 16 + l15];
        c[tc] = (v8f){bn, bn, bn, bn, bn, bn, bn, bn};
    }

    int arow = rowBase + l15;
    if (arow >= N) arow = N - 1;            // clamp (N%16==0 in practice)
    const float* aptr = agg + (size_t)arow * IN_DIM;

#pragma unroll
    for (int kb = 0; kb < IN_DIM / 4; ++kb) {
        const int kbase = kb * 4 + hi * 2;  // even -> 8B-aligned float2 load
        const v2f a = *reinterpret_cast<const v2f*>(aptr + kbase);
        const int kp = kbase >> 1;          // LDS k-pair row
#pragma unroll
        for (int tc = 0; tc < 4; ++tc) {
            const int n = tc * 16 + l15;
            const v2f b = *reinterpret_cast<const v2f*>(&WL[kp * 128 + n * 2]);
            c[tc] = __builtin_amdgcn_wmma_f32_16x16x4_f32(
                        /*neg_a=*/false, a, /*neg_b=*/false, b,
                        /*c_mod=*/(short)0, c[tc],
                        /*reuse_a=*/false, /*reuse_b=*/false);
        }
    }

#pragma unroll
    for (int j = 0; j < 8; ++j) {
        const int row = rowBase + hi * 8 + j;
        if (row < N) {
            float* orow = out + (size_t)row * OUT_DIM + l15;
#pragma unroll
            for (int tc = 0; tc < 4; ++tc) {
                orow[tc * 16] = c[tc][j];
            }
        }
    }
}

// ---------------------------------------------------------------------------
// Launch
// ---------------------------------------------------------------------------
extern "C" void kernel_launch(void* const* d_in, const int* in_sizes, int n_in,
                              void* d_out, int out_size, void* d_ws, size_t ws_size,
                              hipStream_t stream) {
    const float* x    = (const float*)d_in[0];   // [N, 64]
    const int*   ei   = (const int*)d_in[1];     // [2, E]
    const float* W    = (const float*)d_in[2];   // [64, 64]
    const float* bias = (const float*)d_in[3];   // [64]
    float* out = (float*)d_out;

    const int N = in_sizes[0] / IN_DIM;
    const int E = in_sizes[1] / 2;
    const int* src = ei;
    const int* dst = ei + E;

    // Workspace layout: deg_out[N] | deg_in[N] | agg[N*IN_DIM]
    float* ws      = (float*)d_ws;
    float* deg_out = ws;
    float* deg_in  = ws + N;
    float* agg     = ws + 2 * (size_t)N;
    const long zfloats = 2L * N + (long)N * IN_DIM;

    const int TPB = 256;

    // Phase 0: zero workspace
    zero_ws_kernel<<<(unsigned)((zfloats + TPB - 1) / TPB), TPB, 0, stream>>>(ws, zfloats);

    // Phase 1: degrees
    degree_kernel<<<(E + TPB - 1) / TPB, TPB, 0, stream>>>(src, dst, deg_out, deg_in, E);

    // Phase 2: gather + scatter-add (16 lanes per edge)
    const long scatterThreads = 16L * E;
    scatter_kernel<<<(unsigned)((scatterThreads + TPB - 1) / TPB), TPB, 0, stream>>>(
        x, src, dst, deg_out, deg_in, agg, E);

    // Phase 3: WMMA GEMM + bias (one wave per 16-row strip, 4 col tiles each)
    const int rowTiles = (N + 15) / 16;
    const int wavesPerBlock = TPB / 32;
    const int gemmBlocks = (rowTiles + wavesPerBlock - 1) / wavesPerBlock;
    gemm_wmma_kernel<<<gemmBlocks, TPB, 0, stream>>>(agg, W, bias, out, N, rowTiles);
}